// Classification_64295660421696
// MI455X (gfx1250) — compile-verified
//
#include <hip/hip_runtime.h>

#define B_SZ 256
#define N_SZ 2000
#define D_SZ 512
#define C_SZ 100
#define CPAD 112   // 7 tiles of 16

typedef __attribute__((ext_vector_type(2))) float v2f;
typedef __attribute__((ext_vector_type(8))) float v8f;

// --------------------------- CDNA5 async-LDS path ---------------------------
#if defined(__has_builtin)
#if __has_builtin(__builtin_amdgcn_global_load_async_to_lds_b128) && \
    __has_builtin(__builtin_amdgcn_s_wait_asynccnt)
#define HAVE_ASYNC_LDS 1
#endif
#endif

// Builtin signature (from hipcc diagnostic): strict int4-vector pointers,
// AS1 (global) source, AS3 (LDS) destination.
typedef int v4i __attribute__((vector_size(16)));
typedef __attribute__((address_space(1))) v4i as1_v4i;
typedef __attribute__((address_space(3))) v4i as3_v4i;

__device__ static inline as1_v4i* to_glob(const void* p) {
    return (as1_v4i*)(uintptr_t)p;
}
__device__ static inline as3_v4i* to_lds(const void* p) {
    // Flat shared-aperture addr [31:0] == LDS byte address (ISA §10.2 aperture calc).
    return (as3_v4i*)(uint32_t)(uintptr_t)p;
}

// ---------------------------------------------------------------------------
// Stage 0a: zero-pad labels [N][100] -> [N][112] so the WMMA loop is branchless.
// ---------------------------------------------------------------------------
__global__ __launch_bounds__(256)
void pad_labels_kernel(const float* __restrict__ labels, float* __restrict__ labelsP)
{
    int idx = blockIdx.x * 256 + threadIdx.x;
    if (idx < N_SZ * CPAD) {
        int n = idx / CPAD;
        int c = idx - n * CPAD;
        labelsP[idx] = (c < C_SZ) ? labels[n * C_SZ + c] : 0.0f;
    }
}

// Stage 0b: per-class counts (column sums of one-hot), computed once.
__global__ __launch_bounds__(128)
void counts_kernel(const float* __restrict__ labels, float* __restrict__ counts)
{
    int c = threadIdx.x;
    if (c < CPAD) {
        float s = 0.0f;
        if (c < C_SZ)
            for (int n = 0; n < N_SZ; ++n) s += labels[n * C_SZ + c];
        counts[c] = s;
    }
}

// ---------------------------------------------------------------------------
// Stage 1: scores[b][n] = sigmoid( sum_d w[d]*|in[b][d]-sup[n][d]| + bias )
// VALU work; tiles staged into LDS via GLOBAL_LOAD_ASYNC_TO_LDS_B128.
// ---------------------------------------------------------------------------
#define TB 16
#define TN 32
#define SUP_PITCH 516   // 512 + 4 pad floats: 16B-aligned rows, breaks bank aliasing

__global__ __launch_bounds__(256)
void scores_kernel(const float* __restrict__ inputs,
                   const float* __restrict__ support,
                   const float* __restrict__ w,
                   const float* __restrict__ bias,
                   float* __restrict__ scores)
{
    __shared__ __align__(16) float sSup[TN * SUP_PITCH];   // ~66 KB
    __shared__ __align__(16) float sIn[TB * D_SZ];         // 32 KB
    __shared__ __align__(16) float sW[D_SZ];               // 2 KB

    const int tid = threadIdx.x;
    const int n0  = blockIdx.x * TN;
    const int b0  = blockIdx.y * TB;

#if HAVE_ASYNC_LDS
    // Async 16B/lane global->LDS copies, tracked by ASYNCcnt (no VGPR round trip).
    for (int idx4 = tid; idx4 < (TN * D_SZ) / 4; idx4 += 256) {
        int idx = idx4 * 4;
        int r = idx >> 9;          // / 512
        int c = idx & 511;
        int n = n0 + r;
        if (n >= N_SZ) n = N_SZ - 1;   // clamp: rows past N are computed but never stored
        __builtin_amdgcn_global_load_async_to_lds_b128(
            to_glob(support + n * D_SZ + c), to_lds(&sSup[r * SUP_PITCH + c]), 0, 0);
    }
    for (int idx4 = tid; idx4 < (TB * D_SZ) / 4; idx4 += 256) {
        int idx = idx4 * 4;
        __builtin_amdgcn_global_load_async_to_lds_b128(
            to_glob(inputs + b0 * D_SZ + idx), to_lds(&sIn[idx]), 0, 0);
    }
    if (tid < D_SZ / 4) {
        __builtin_amdgcn_global_load_async_to_lds_b128(
            to_glob(w + tid * 4), to_lds(&sW[tid * 4]), 0, 0);
    }
    __builtin_amdgcn_s_wait_asynccnt(0);
    __syncthreads();
#else
    for (int idx = tid; idx < TN * D_SZ; idx += 256) {
        int r = idx >> 9;
        int c = idx & 511;
        int n = n0 + r;
        sSup[r * SUP_PITCH + c] = (n < N_SZ) ? support[n * D_SZ + c] : 0.0f;
    }
    for (int idx = tid; idx < TB * D_SZ; idx += 256)
        sIn[idx] = inputs[b0 * D_SZ + idx];
    for (int idx = tid; idx < D_SZ; idx += 256)
        sW[idx] = w[idx];
    __syncthreads();
#endif

    const int nl = tid & 31;   // lane = local n (bank-spread via pitch)
    const int bg = tid >> 5;   // wave id -> owns 2 b rows (broadcast LDS reads)
    const float bia = bias[0];

    const float4* supRow = (const float4*)(&sSup[nl * SUP_PITCH]);
    const float4* inRow0 = (const float4*)(&sIn[(bg * 2 + 0) * D_SZ]);
    const float4* inRow1 = (const float4*)(&sIn[(bg * 2 + 1) * D_SZ]);
    const float4* wRow   = (const float4*)(sW);

    float acc0 = 0.0f, acc1 = 0.0f;
    #pragma unroll 4
    for (int d4 = 0; d4 < D_SZ / 4; ++d4) {
        float4 s  = supRow[d4];
        float4 wv = wRow[d4];
        float4 a0 = inRow0[d4];
        float4 a1 = inRow1[d4];
        acc0 += wv.x * fabsf(a0.x - s.x) + wv.y * fabsf(a0.y - s.y)
              + wv.z * fabsf(a0.z - s.z) + wv.w * fabsf(a0.w - s.w);
        acc1 += wv.x * fabsf(a1.x - s.x) + wv.y * fabsf(a1.y - s.y)
              + wv.z * fabsf(a1.z - s.z) + wv.w * fabsf(a1.w - s.w);
    }

    const int n = n0 + nl;
    if (n < N_SZ) {
        float s0 = 1.0f / (1.0f + __expf(-(acc0 + bia)));
        float s1 = 1.0f / (1.0f + __expf(-(acc1 + bia)));
        scores[(b0 + bg * 2 + 0) * N_SZ + n] = s0;   // coalesced over lanes
        scores[(b0 + bg * 2 + 1) * N_SZ + n] = s1;
    }
}

// ---------------------------------------------------------------------------
// Stage 2: out = divide_no_nan(scores @ labels, counts) on V_WMMA_F32_16X16X4_F32.
// 4 waves split K=2000 (stride 16); constant trip count (125 = 5*25) so the
// backend fully unrolls 5 WMMAs per body with clause-able constant offsets.
// ---------------------------------------------------------------------------
__global__ __launch_bounds__(128)
void class_kernel(const float* __restrict__ scores,
                  const float* __restrict__ labelsP,   // [N][CPAD] zero-padded
                  const float* __restrict__ counts,    // [CPAD]
                  float* __restrict__ out)
{
    __shared__ float red[4 * 8 * 32];

    const int lane = threadIdx.x & 31;
    const int wid  = threadIdx.x >> 5;
    const int b0   = blockIdx.x * 16;
    const int c0   = blockIdx.y * 16;

    const int m    = lane & 15;
    const int half = lane >> 4;        // ISA 16x4 A / 4x16 B: halves hold K pairs
    const int col  = c0 + m;

    v8f acc = {0.f, 0.f, 0.f, 0.f, 0.f, 0.f, 0.f, 0.f};
    const float* arow = scores  + (b0 + m) * N_SZ + wid * 4 + 2 * half;
    const float* brow = labelsP + (wid * 4 + 2 * half) * CPAD + col;

    #pragma unroll 5
    for (int i = 0; i < N_SZ / 16; ++i) {   // 125 iterations, compile-time bound
        const int k = i * 16;
        v2f a, b;
        a[0] = arow[k];                  // A[m][k'], A[m][k'+1]   (k' = k+wid*4+2h)
        a[1] = arow[k + 1];
        b[0] = brow[(k    ) * CPAD];     // B[k'][col], B[k'+1][col]
        b[1] = brow[(k + 1) * CPAD];
        acc = __builtin_amdgcn_wmma_f32_16x16x4_f32(
                  /*neg_a=*/false, a, /*neg_b=*/false, b,
                  /*c_mod=*/(short)0, acc, /*reuse_a=*/false, /*reuse_b=*/false);
    }

    #pragma unroll
    for (int r = 0; r < 8; ++r)
        red[(wid * 8 + r) * 32 + lane] = acc[r];
    __syncthreads();

    if (wid == 0) {
        float cnt = counts[col];
        float inv = (cnt != 0.0f) ? (1.0f / cnt) : 0.0f;
        #pragma unroll
        for (int r = 0; r < 8; ++r) {
            // Deterministic fixed-order reduction across the 4 waves.
            float s = red[(0 * 8 + r) * 32 + lane]
                    + red[(1 * 8 + r) * 32 + lane]
                    + red[(2 * 8 + r) * 32 + lane]
                    + red[(3 * 8 + r) * 32 + lane];
            int row = b0 + half * 8 + r;   // C/D layout: halves = M 0-7 / 8-15
            if (col < C_SZ)
                out[row * C_SZ + col] = s * inv;
        }
    }
}

// ---------------------------------------------------------------------------
extern "C" void kernel_launch(void* const* d_in, const int* in_sizes, int n_in,
                              void* d_out, int out_size, void* d_ws, size_t ws_size,
                              hipStream_t stream)
{
    (void)in_sizes; (void)n_in; (void)out_size; (void)ws_size;

    const float* inputs  = (const float*)d_in[0];  // [B, D]
    const float* support = (const float*)d_in[1];  // [N, D]
    const float* labels  = (const float*)d_in[2];  // [N, C]
    const float* w       = (const float*)d_in[3];  // [D]
    const float* bias    = (const float*)d_in[4];  // scalar
    float* out = (float*)d_out;                    // [B, C]

    // Workspace layout (floats): scores[B*N] | labelsP[N*CPAD] | counts[CPAD]
    float* wsf     = (float*)d_ws;
    float* scores  = wsf;                          // 512000 f = 2,048,000 B
    float* labelsP = scores + B_SZ * N_SZ;         // 224000 f =   896,000 B
    float* counts  = labelsP + N_SZ * CPAD;        //    112 f

    pad_labels_kernel<<<(N_SZ * CPAD + 255) / 256, 256, 0, stream>>>(labels, labelsP);
    counts_kernel<<<1, 128, 0, stream>>>(labels, counts);

    dim3 g1((N_SZ + TN - 1) / TN, B_SZ / TB);      // 63 x 16 blocks, 256 thr
    scores_kernel<<<g1, 256, 0, stream>>>(inputs, support, w, bias, scores);

    dim3 g2(B_SZ / 16, CPAD / 16);                 // 16 x 7 blocks, 128 thr
    class_kernel<<<g2, 128, 0, stream>>>(scores, labelsP, counts, out);
}